// GNN_Layer3_34832184770736
// MI455X (gfx1250) — compile-verified
//
#include <hip/hip_runtime.h>
#include <hip/hip_bf16.h>

// ---------------------------------------------------------------------------
// GNN layer for MI455X (gfx1250, wave32, WMMA).
// fp32 WMMA (V_WMMA_F32_16X16X4_F32): workload is memory-bound (118 GFLOP vs
// ~2 GB HBM traffic @ 23.3 TB/s), so full-precision matrix ops are free.
// hu/hw/sums (77 MB) are L2-resident (192 MB L2) -> gathers/atomics stay on chip.
// messages are never written to HBM (fused edge kernel).
// ---------------------------------------------------------------------------

typedef __attribute__((ext_vector_type(2))) float v2f;
typedef __attribute__((ext_vector_type(8))) float v8f;

#define N_NODES   50000
#define IN_DIM    256
#define EDGE_DIM  128
#define OUT_DIM   128
#define N_EDGES   1600000

__device__ __forceinline__ v8f wmma_f32(v2f a, v2f b, v8f c) {
  // 8 args: (neg_a, A, neg_b, B, c_mod, C, reuse_a, reuse_b)
  return __builtin_amdgcn_wmma_f32_16x16x4_f32(
      false, a, false, b, (short)0, c, false, false);
}

__device__ __forceinline__ float leaky(float v) {
  return v >= 0.0f ? v : 0.01f * v;
}

// ---------------------------------------------------------------------------
// Kernel 0: zero sums (+counts) region of workspace
// ---------------------------------------------------------------------------
__global__ void zero_kernel(float4* __restrict__ p, int n4) {
  int i = blockIdx.x * blockDim.x + threadIdx.x;
  int stride = gridDim.x * blockDim.x;
  const float4 z = make_float4(0.f, 0.f, 0.f, 0.f);
  for (; i < n4; i += stride) p[i] = z;
}

// ---------------------------------------------------------------------------
// Kernel 1: node projections. hu = x W_hu^T, hw = x W_hw^T, z2 = x W2^T,
// zemb = x W_emb^T.  One block = 16 nodes; 4 waves each own one weight.
// ---------------------------------------------------------------------------
#define XPAD 260   // 256 + 4-float pad -> conflict-free ds_load_b64 per row

__global__ __launch_bounds__(128) void node_proj_kernel(
    const float* __restrict__ x,
    const float* __restrict__ W_hu, const float* __restrict__ W_hw,
    const float* __restrict__ W2,   const float* __restrict__ W_emb,
    float* __restrict__ hu, float* __restrict__ hw,
    float* __restrict__ z2, float* __restrict__ zemb)
{
  __shared__ float xs[16 * XPAD];
  const int node0 = blockIdx.x * 16;
  const int tid = threadIdx.x;

  // cooperative load of 16x256 x-tile (float4, 16B aligned: XPAD*4 % 16 == 0)
  for (int i = tid; i < 16 * (IN_DIM / 4); i += 128) {
    const int row = i >> 6, c4 = i & 63;
    const float4 v = *(const float4*)(x + (node0 + row) * IN_DIM + c4 * 4);
    *(float4*)(&xs[row * XPAD + c4 * 4]) = v;
  }
  __syncthreads();

  const int wave = tid >> 5;
  const int lane = tid & 31;
  const float* W = (wave == 0) ? W_hu : (wave == 1) ? W_hw : (wave == 2) ? W2 : W_emb;
  float* out     = (wave == 0) ? hu   : (wave == 1) ? hw   : (wave == 2) ? z2 : zemb;

  const int m    = lane & 15;   // A row / B col / D col
  const int hi   = lane >> 4;   // lane-half selects K+0 / K+2
  const int ksel = hi * 2;

  v8f acc[8] = {};
  for (int k0 = 0; k0 < IN_DIM; k0 += 4) {
    const v2f a = *(const v2f*)(&xs[m * XPAD + k0 + ksel]);
#pragma unroll
    for (int nt = 0; nt < 8; ++nt) {
      const v2f b = *(const v2f*)(W + (nt * 16 + m) * IN_DIM + k0 + ksel);
      acc[nt] = wmma_f32(a, b, acc[nt]);
    }
  }
  // D layout: VGPR r -> rows r (lanes 0-15) and r+8 (lanes 16-31), col = m
#pragma unroll
  for (int nt = 0; nt < 8; ++nt) {
#pragma unroll
    for (int r = 0; r < 8; ++r) {
      const int M = r + hi * 8;
      out[(node0 + M) * OUT_DIM + nt * 16 + m] = acc[nt][r];
    }
  }
}

// ---------------------------------------------------------------------------
// Kernel 2: fused edge kernel.  Per 16-edge tile (1 wave):
//   pre = edge_attr W_e^T                     (WMMA GEMM 1)
//   msg = leaky(pre + hu[src] + hw[tgt])      (L2-resident gathers)
//   sums[tgt] += msg ; counts[tgt] += 1       (L2 atomics)
//   attributes = msg W_attr^T                 (WMMA GEMM 2, via LDS relayout)
// ---------------------------------------------------------------------------
#define WPB  4     // waves per block
#define MPAD 132   // 128 + 4-float pad

__global__ __launch_bounds__(32 * WPB) void edge_kernel(
    const float* __restrict__ edge_attr,
    const long long* __restrict__ eidx,      // int64 [2][E]
    const float* __restrict__ W_e, const float* __restrict__ W_attr,
    const float* __restrict__ hu,  const float* __restrict__ hw,
    float* __restrict__ sums, float* __restrict__ counts,
    float* __restrict__ attrs_out)
{
  __shared__ float ms[WPB][16 * MPAD];
  __shared__ int   sid[WPB][16];
  __shared__ int   tgt[WPB][16];

  const int tid  = threadIdx.x;
  const int wave = tid >> 5;
  const int lane = tid & 31;
  const int e0   = (blockIdx.x * WPB + wave) * 16;

  const int m    = lane & 15;
  const int hi   = lane >> 4;
  const int ksel = hi * 2;

  if (lane < 16) {
    sid[wave][lane] = (int)eidx[e0 + lane];
    tgt[wave][lane] = (int)eidx[N_EDGES + e0 + lane];
  }
  __syncthreads();

  // GEMM 1: pre-activation messages (A = edge_attr tile, direct from global)
  v8f acc[8] = {};
  const float* Aptr = edge_attr + (long long)(e0 + m) * EDGE_DIM;
  for (int k0 = 0; k0 < EDGE_DIM; k0 += 4) {
    const v2f a = *(const v2f*)(Aptr + k0 + ksel);
#pragma unroll
    for (int nt = 0; nt < 8; ++nt) {
      const v2f b = *(const v2f*)(W_e + (nt * 16 + m) * EDGE_DIM + k0 + ksel);
      acc[nt] = wmma_f32(a, b, acc[nt]);
    }
  }

  // gathers + leaky, stage to LDS (D-layout -> row major), scatter atomics
#pragma unroll
  for (int nt = 0; nt < 8; ++nt) {
    const int n = nt * 16 + m;
#pragma unroll
    for (int r = 0; r < 8; ++r) {
      const int M = r + hi * 8;
      const int s = sid[wave][M];
      const int t = tgt[wave][M];
      float v = acc[nt][r] + hu[s * OUT_DIM + n] + hw[t * OUT_DIM + n];
      v = leaky(v);
      ms[wave][M * MPAD + n] = v;
      atomicAdd(&sums[t * OUT_DIM + n], v);   // no-return global_atomic_add_f32
    }
  }
  if (lane < 16) atomicAdd(&counts[tgt[wave][lane]], 1.0f);

  // GEMM 2: attributes = msg W_attr^T (A re-read from LDS in A-layout;
  // same-wave DS ops are in-order so the tile is visible without a barrier)
  v8f acc2[8] = {};
  for (int k0 = 0; k0 < OUT_DIM; k0 += 4) {
    const v2f a = *(const v2f*)(&ms[wave][m * MPAD + k0 + ksel]);
#pragma unroll
    for (int nt = 0; nt < 8; ++nt) {
      const v2f b = *(const v2f*)(W_attr + (nt * 16 + m) * OUT_DIM + k0 + ksel);
      acc2[nt] = wmma_f32(a, b, acc2[nt]);
    }
  }
#pragma unroll
  for (int nt = 0; nt < 8; ++nt) {
#pragma unroll
    for (int r = 0; r < 8; ++r) {
      const int M = r + hi * 8;
      attrs_out[(long long)(e0 + M) * OUT_DIM + nt * 16 + m] = acc2[nt][r];
    }
  }
}

// ---------------------------------------------------------------------------
// Kernel 3: embeddings = zemb + leaky(z2 + sums / max(counts, 1))
// ---------------------------------------------------------------------------
__global__ void finalize_kernel(
    const float* __restrict__ z2, const float* __restrict__ zemb,
    const float* __restrict__ sums, const float* __restrict__ counts,
    float* __restrict__ emb_out)
{
  const int i = blockIdx.x * blockDim.x + threadIdx.x;
  if (i >= N_NODES * OUT_DIM) return;
  const int node = i >> 7;   // OUT_DIM == 128
  const float denom = fmaxf(counts[node], 1.0f);
  const float agg = sums[i] / denom;
  emb_out[i] = zemb[i] + leaky(z2[i] + agg);
}

// ---------------------------------------------------------------------------
extern "C" void kernel_launch(void* const* d_in, const int* in_sizes, int n_in,
                              void* d_out, int out_size, void* d_ws, size_t ws_size,
                              hipStream_t stream) {
  (void)in_sizes; (void)n_in; (void)out_size; (void)ws_size;

  const float*     x         = (const float*)d_in[0];
  const long long* eidx      = (const long long*)d_in[1];  // int64 [2][E]
  const float*     edge_attr = (const float*)d_in[2];
  const float*     W_e       = (const float*)d_in[3];
  const float*     W_hu      = (const float*)d_in[4];
  const float*     W_hw      = (const float*)d_in[5];
  const float*     W2        = (const float*)d_in[6];
  const float*     W_emb     = (const float*)d_in[7];
  const float*     W_attr    = (const float*)d_in[8];

  // workspace layout (floats)
  float* ws     = (float*)d_ws;
  float* hu     = ws;                        // 6.4M
  float* hw     = ws + 6400000;              // 6.4M
  float* z2     = ws + 12800000;             // 6.4M
  float* zemb   = ws + 19200000;             // 6.4M
  float* sums   = ws + 25600000;             // 6.4M
  float* counts = ws + 32000000;             // 50k   (contiguous with sums)

  float* emb_out   = (float*)d_out;          // 50000 x 128
  float* attrs_out = emb_out + N_NODES * OUT_DIM;  // 1.6M x 128

  // 0) reset accumulators (sums + counts = 6,450,000 floats, /4 exact)
  zero_kernel<<<1024, 256, 0, stream>>>((float4*)sums, (6400000 + N_NODES) / 4);

  // 1) node projections: 50000/16 = 3125 tiles
  node_proj_kernel<<<3125, 128, 0, stream>>>(x, W_hu, W_hw, W2, W_emb,
                                             hu, hw, z2, zemb);

  // 2) fused edge kernel: 1.6M / (16 * WPB) = 25000 blocks
  edge_kernel<<<N_EDGES / (16 * WPB), 32 * WPB, 0, stream>>>(
      edge_attr, eidx, W_e, W_attr, hu, hw, sums, counts, attrs_out);

  // 3) finalize embeddings: 6.4M elements
  finalize_kernel<<<(N_NODES * OUT_DIM) / 256, 256, 0, stream>>>(
      z2, zemb, sums, counts, emb_out);
}